// InterDistanceLossV2_5179730559713
// MI455X (gfx1250) — compile-verified
//
#include <hip/hip_runtime.h>

typedef float v2f __attribute__((ext_vector_type(2)));
typedef float v8f __attribute__((ext_vector_type(8)));

#define BATCH 32
#define NPTS  1024
#define DIM   3
#define TILES (NPTS / 16)   // 64 column tiles per strip
#define EPSV  1e-8f

// One WMMA f32 16x16x4 operand element pair for this lane plus the FULL
// squared norm of the loaded point (all 3 components come from one b96 load).
// A/B f32 K=4 layout (ISA 7.12.2): lanes 0-15 hold (K=0,K=1) in (v0,v1),
// lanes 16-31 hold (K=2,K=3). D=3 -> K=3 padded with zero.
struct OpLoad {
    v2f   op;
    float sqn;
};

__device__ inline OpLoad load_op(const float* __restrict__ base, int row, bool hi) {
    const float* p = base + (size_t)row * DIM;
    float c0 = p[0], c1 = p[1], c2 = p[2];
    OpLoad r;
    r.op.x = hi ? c2  : c0;
    r.op.y = hi ? 0.f : c1;
    r.sqn  = fmaf(c0, c0, fmaf(c1, c1, c2 * c2));
    return r;
}

__global__ void idl_init_kernel(float* ws) {
    if (threadIdx.x == 0) { ws[0] = 0.f; ws[1] = 0.f; }
}

__global__ __launch_bounds__(256)
void idl_main_kernel(const float* __restrict__ preds,
                     const float* __restrict__ targets,
                     const int*   __restrict__ mask,
                     float*       __restrict__ ws) {
    const int tid   = threadIdx.x;
    const int lane  = tid & 31;
    const int wave  = tid >> 5;
    const int strip = blockIdx.x * 8 + wave;          // 2048 strips total
    const int batch = strip >> 6;                      // / TILES
    const int ti    = strip & (TILES - 1);
    const int m0    = ti * 16;
    const bool hi   = lane >= 16;
    const int  lrow = lane & 15;
    const int  hi8  = hi ? 8 : 0;

    const float* pB = preds   + (size_t)batch * NPTS * DIM;
    const float* tB = targets + (size_t)batch * NPTS * DIM;
    const int*   mB = mask    + (size_t)batch * NPTS;

    // A operands: rows m0..m0+15 of preds/targets (full norms come for free)
    OpLoad Ap = load_op(pB, m0 + lrow, hi);
    OpLoad At = load_op(tB, m0 + lrow, hi);

    // Broadcast row norms / masks to match C/D layout:
    // VGPR r holds row M = r + 8*(lane>=16). Prologue-only shuffles.
    float rnp[8], rnt[8], mm[8];
#pragma unroll
    for (int r = 0; r < 8; ++r) {
        rnp[r] = __shfl(Ap.sqn, r + hi8, 32);
        rnt[r] = __shfl(At.sqn, r + hi8, 32);
        mm[r]  = (float)mB[m0 + r + hi8];
    }

    float num = 0.f, den = 0.f;

    // Software-pipelined over column tiles: prefetch j+1 while computing j.
    OpLoad Bp = load_op(pB, lrow, hi);
    OpLoad Bt = load_op(tB, lrow, hi);
    float  mn = (float)mB[lrow];

    for (int j = 0; j < TILES; ++j) {
        const int jn  = (j + 1) & (TILES - 1);   // wrapped: branch-free prefetch
        const int n0n = jn * 16;
        OpLoad nBp = load_op(pB, n0n + lrow, hi);
        OpLoad nBt = load_op(tB, n0n + lrow, hi);
        float  nmn = (float)mB[n0n + lrow];

        // Gram tiles: D = A(16x4) x B(4x16) for preds and targets
        v8f cz = {};
        v8f gp = __builtin_amdgcn_wmma_f32_16x16x4_f32(
            false, Ap.op, false, Bp.op, (short)0, cz, false, false);
        v8f gt = __builtin_amdgcn_wmma_f32_16x16x4_f32(
            false, At.op, false, Bt.op, (short)0, cz, false, false);

        const float cnp = Bp.sqn;   // col norm for N = lane&15 (local, no shuffle)
        const float cnt = Bt.sqn;
        const float mnc = mn;

#pragma unroll
        for (int r = 0; r < 8; ++r) {
            float sqp = rnp[r] + cnp - 2.f * gp[r];
            float sqt = rnt[r] + cnt - 2.f * gt[r];
            float dp  = sqp > 0.f ? __builtin_amdgcn_sqrtf(sqp) : 0.f;
            float dt  = sqt > 0.f ? __builtin_amdgcn_sqrtf(sqt) : 0.f;
            float w   = mm[r] * mnc;
            num = fmaf(fabsf(dp - dt) * __builtin_amdgcn_rcpf(dt + EPSV), w, num);
            den += w;
        }

        Bp = nBp; Bt = nBt; mn = nmn;
    }

    // Wave32 tree reduction
#pragma unroll
    for (int s = 16; s >= 1; s >>= 1) {
        num += __shfl_xor(num, s, 32);
        den += __shfl_xor(den, s, 32);
    }
    if (lane == 0) {
        atomicAdd(&ws[0], num);
        atomicAdd(&ws[1], den);
    }
}

__global__ void idl_final_kernel(const float* __restrict__ ws,
                                 float* __restrict__ out) {
    if (threadIdx.x == 0) out[0] = ws[0] / ws[1];
}

extern "C" void kernel_launch(void* const* d_in, const int* in_sizes, int n_in,
                              void* d_out, int out_size, void* d_ws, size_t ws_size,
                              hipStream_t stream) {
    const float* preds   = (const float*)d_in[0];
    const float* targets = (const float*)d_in[1];
    const int*   mask    = (const int*)d_in[2];
    float* out = (float*)d_out;
    float* ws  = (float*)d_ws;

    idl_init_kernel<<<1, 32, 0, stream>>>(ws);
    // 32 batches x 64 row-tiles = 2048 strips; 8 waves per 256-thread block
    idl_main_kernel<<<(BATCH * TILES) / 8, 256, 0, stream>>>(preds, targets, mask, ws);
    idl_final_kernel<<<1, 32, 0, stream>>>(ws, out);
}